// VectorQuantizer_62560493633541
// MI455X (gfx1250) — compile-verified
//
#include <hip/hip_runtime.h>
#include <hip/hip_bf16.h>

typedef __attribute__((ext_vector_type(2))) float v2f;
typedef __attribute__((ext_vector_type(4))) float v4f;
typedef __attribute__((ext_vector_type(8))) float v8f;
typedef __attribute__((ext_vector_type(4))) int   v4i;

typedef __attribute__((address_space(1))) v4i* gasync_ptr;   // global src (AS1)
typedef __attribute__((address_space(3))) v4i* lasync_ptr;   // LDS dst (AS3)

#define N_TOK   32768
#define K_CODES 4096
#define DIM     256
#define TILE_N  16
#define WAVES   8
#define ROWS_PER_BLOCK 128   // 8 waves * 16 rows
#define LDS_STRIDE 260       // padded: conflict-free ds_load_b64 B fragments
#define NEG_INF (-3.402823466e38f)

#if defined(__has_builtin)
#if __has_builtin(__builtin_amdgcn_global_load_async_to_lds_b128)
#define VQ_ASYNC_LDS 1
#endif
#endif

#if defined(__has_builtin)
#if __has_builtin(__builtin_amdgcn_s_wait_asynccnt)
#define WAIT_ASYNC(n) __builtin_amdgcn_s_wait_asynccnt(n)
#else
#define WAIT_ASYNC(n) asm volatile("s_wait_asynccnt %0" ::"i"(n) : "memory")
#endif
#endif

// ---------------------------------------------------------------------------
// Kernel 1: bias[k] = -0.5 * ||w_k||^2  (argmin d2 == argmax (x.w + bias))
// Also zeroes the loss accumulator.
// ---------------------------------------------------------------------------
__global__ __launch_bounds__(256) void vq_bias_kernel(const float* __restrict__ W,
                                                      float* __restrict__ bias,
                                                      float* __restrict__ acc) {
    int gwave = (blockIdx.x * blockDim.x + threadIdx.x) >> 5;
    int lane  = threadIdx.x & 31;
    if (gwave < K_CODES) {
        const float* w = W + gwave * DIM;
        float s = 0.f;
        #pragma unroll
        for (int j = 0; j < 8; ++j) {
            float x = w[lane + 32 * j];
            s += x * x;
        }
        #pragma unroll
        for (int off = 16; off > 0; off >>= 1)
            s += __shfl_xor(s, off, 32);
        if (lane == 0) bias[gwave] = -0.5f * s;
    }
    if (blockIdx.x == 0 && threadIdx.x == 0) *acc = 0.f;
}

// ---------------------------------------------------------------------------
// Inner MMA over one 16-code tile: two independent WMMA chains (even/odd k)
// with the B fragment pipeline one pair ahead of the consuming WMMAs.
// ---------------------------------------------------------------------------
__device__ __forceinline__ void vq_tile_mma(const float* __restrict__ brow,
                                            const v2f (&a)[64], v8f& ca, v8f& cb) {
    v2f b0 = *(const v2f*)(brow + 0);
    v2f b1 = *(const v2f*)(brow + 4);
    #pragma unroll
    for (int kk = 0; kk < 64; kk += 2) {
        v2f bn0 = b0, bn1 = b1;
        if (kk + 2 < 64) {
            bn0 = *(const v2f*)(brow + (kk + 2) * 4);
            bn1 = *(const v2f*)(brow + (kk + 3) * 4);
        }
        ca = __builtin_amdgcn_wmma_f32_16x16x4_f32(
                 false, a[kk],     false, b0, (short)0, ca, false, false);
        cb = __builtin_amdgcn_wmma_f32_16x16x4_f32(
                 false, a[kk + 1], false, b1, (short)0, cb, false, false);
        b0 = bn0; b1 = bn1;
    }
}

// ---------------------------------------------------------------------------
// Kernel 2: fused GEMM + running argmax via V_WMMA_F32_16X16X4_F32.
// score[n][k] = x_n . w_k - 0.5||w_k||^2 ; idx[n] = argmax_k score.
// Codebook tiles double-buffered in LDS via GLOBAL_LOAD_ASYNC_TO_LDS_B128.
// ---------------------------------------------------------------------------
__global__ __launch_bounds__(256) void vq_argmax_kernel(const float* __restrict__ X,
                                                        const float* __restrict__ W,
                                                        const float* __restrict__ bias,
                                                        int* __restrict__ idx) {
#ifdef VQ_ASYNC_LDS
    __shared__ float ldsB[2][TILE_N * LDS_STRIDE];   // 2 x 16.25 KB tiles
#else
    __shared__ float ldsB[1][TILE_N * LDS_STRIDE];
#endif

    const int t    = threadIdx.x;
    const int lane = t & 31;
    const int wave = t >> 5;
    const int n16  = lane & 15;        // C-matrix column within tile / A row
    const int half = lane >> 4;        // K-half selector for A/B fragments
    const int rowBase = blockIdx.x * ROWS_PER_BLOCK + wave * TILE_N;

    // --- Load this wave's 16x256 A-panel into registers (read HBM once). ---
    // A 16x4 f32 fragment layout: lanes 0-15 M=0..15 {K0,K1}; lanes 16-31 {K2,K3}.
    v2f a[64];
    const float* xrow = X + (size_t)(rowBase + n16) * DIM + 2 * half;
    #pragma unroll
    for (int kk = 0; kk < 64; ++kk)
        a[kk] = *(const v2f*)(xrow + kk * 4);

    float best[8];
    int   bcol[8];
    #pragma unroll
    for (int v = 0; v < 8; ++v) { best[v] = NEG_INF; bcol[v] = 0; }

    const int cc  = t >> 4;            // cooperative-load: code row 0..15
    const int l16 = t & 15;            // 16 threads per code row

#ifdef VQ_ASYNC_LDS
    // ---------------- async double-buffered pipeline -----------------------
    {
        const float* src0 = W + (size_t)cc * DIM + l16 * 4;
        float*       dst0 = &ldsB[0][cc * LDS_STRIDE + l16 * 4];
        #pragma unroll
        for (int i = 0; i < 4; ++i)
            __builtin_amdgcn_global_load_async_to_lds_b128(
                (gasync_ptr)(float*)(src0 + i * 64),
                (lasync_ptr)(dst0 + i * 64), 0, 0);
    }
    int p = 0;
    for (int cb = 0; cb < K_CODES; cb += TILE_N) {
        if (cb + TILE_N < K_CODES) {
            // Stream next tile into the other buffer while this one computes.
            const float* src = W + (size_t)(cb + TILE_N + cc) * DIM + l16 * 4;
            float*       dst = &ldsB[p ^ 1][cc * LDS_STRIDE + l16 * 4];
            #pragma unroll
            for (int i = 0; i < 4; ++i)
                __builtin_amdgcn_global_load_async_to_lds_b128(
                    (gasync_ptr)(float*)(src + i * 64),
                    (lasync_ptr)(dst + i * 64), 0, 0);
            WAIT_ASYNC(4);   // in-order: current tile's 4 copies retired
        } else {
            WAIT_ASYNC(0);
        }
        __syncthreads();

        float bv = bias[cb + n16];
        v8f ca, cz;
        #pragma unroll
        for (int v = 0; v < 8; ++v) { ca[v] = bv; cz[v] = 0.f; }

        // B 4x16 f32 fragment: lanes 0-15 N=0..15 {K0,K1}; lanes 16-31 {K2,K3}.
        vq_tile_mma(&ldsB[p][n16 * LDS_STRIDE + 2 * half], a, ca, cz);

        #pragma unroll
        for (int v = 0; v < 8; ++v) {
            float s = ca[v] + cz[v];
            if (s > best[v]) { best[v] = s; bcol[v] = cb + n16; }
        }
        __syncthreads();   // protect ldsB[p] before it becomes the async target
        p ^= 1;
    }
#else
    // ---------------- synchronous fallback --------------------------------
    for (int cb = 0; cb < K_CODES; cb += TILE_N) {
        {
            const float* src = W + (size_t)(cb + cc) * DIM + l16 * 4;
            float*       dst = &ldsB[0][cc * LDS_STRIDE + l16 * 4];
            #pragma unroll
            for (int i = 0; i < 4; ++i) {
                v4f v = *(const v4f*)(src + i * 64);
                *(v4f*)(dst + i * 64) = v;
            }
            if (cb + TILE_N < K_CODES)
                __builtin_prefetch(src + TILE_N * DIM, 0, 0);
        }
        __syncthreads();

        float bv = bias[cb + n16];
        v8f ca, cz;
        #pragma unroll
        for (int v = 0; v < 8; ++v) { ca[v] = bv; cz[v] = 0.f; }

        vq_tile_mma(&ldsB[0][n16 * LDS_STRIDE + 2 * half], a, ca, cz);

        #pragma unroll
        for (int v = 0; v < 8; ++v) {
            float s = ca[v] + cz[v];
            if (s > best[v]) { best[v] = s; bcol[v] = cb + n16; }
        }
        __syncthreads();
    }
#endif

    // --- Reduce across the 16 lanes sharing each row group (xor stays in half). ---
    #pragma unroll
    for (int off = 1; off < 16; off <<= 1) {
        #pragma unroll
        for (int v = 0; v < 8; ++v) {
            float ob = __shfl_xor(best[v], off, 32);
            int   oc = __shfl_xor(bcol[v], off, 32);
            if (ob > best[v] || (ob == best[v] && oc < bcol[v])) {
                best[v] = ob; bcol[v] = oc;
            }
        }
    }
    // C layout: VGPR v -> row = v + 8*half, col = n16. Lanes 0 and 16 write.
    if (n16 == 0) {
        #pragma unroll
        for (int v = 0; v < 8; ++v)
            idx[rowBase + half * 8 + v] = bcol[v];
    }
}

// ---------------------------------------------------------------------------
// Kernel 3: gather quantized rows + partial MSE sum (one f32 atomic / block).
// ---------------------------------------------------------------------------
__global__ __launch_bounds__(256) void vq_gather_kernel(const float* __restrict__ X,
                                                        const float* __restrict__ W,
                                                        const int* __restrict__ idx,
                                                        float* __restrict__ out,
                                                        float* __restrict__ acc) {
    __shared__ float partial[WAVES];
    int lane = threadIdx.x & 31;
    int wave = threadIdx.x >> 5;
    int r    = blockIdx.x * WAVES + wave;          // one row per wave

    const float* wr = W + (size_t)idx[r] * DIM;
    const float* xr = X + (size_t)r * DIM;
    float*       o  = out + (size_t)r * DIM;

    float s = 0.f;
    #pragma unroll
    for (int j = 0; j < 2; ++j) {
        int d = lane * 4 + j * 128;
        v4f q = *(const v4f*)(wr + d);
        v4f x = *(const v4f*)(xr + d);
        *(v4f*)(o + d) = q;                         // straight-through fwd == quantized
        v4f df = q - x;
        s += df[0]*df[0] + df[1]*df[1] + df[2]*df[2] + df[3]*df[3];
    }
    #pragma unroll
    for (int off = 16; off > 0; off >>= 1)
        s += __shfl_xor(s, off, 32);
    if (lane == 0) partial[wave] = s;
    __syncthreads();
    if (threadIdx.x == 0) {
        float blk = 0.f;
        #pragma unroll
        for (int w = 0; w < WAVES; ++w) blk += partial[w];
        atomicAdd(acc, blk);
    }
}

// loss = (DIVERGENCE_COST + COMMITMENT_COST) * mean((q - x)^2) = 1.25 * MSE
__global__ void vq_finalize_kernel(const float* __restrict__ acc, float* __restrict__ loss) {
    *loss = 1.25f * (*acc) * (1.0f / (float)(N_TOK * DIM));
}

// ---------------------------------------------------------------------------
extern "C" void kernel_launch(void* const* d_in, const int* in_sizes, int n_in,
                              void* d_out, int out_size, void* d_ws, size_t ws_size,
                              hipStream_t stream) {
    const float* X = (const float*)d_in[0];   // [32768, 256]
    const float* W = (const float*)d_in[1];   // [4096, 256]
    float* out     = (float*)d_out;           // [32768*256] quantized_st, then [1] loss

    float* bias = (float*)d_ws;               // [4096]
    float* acc  = bias + K_CODES;             // [1]
    int*   idx  = (int*)((float*)d_ws + 4352);// [32768], aligned past bias+acc

    vq_bias_kernel  <<<(K_CODES * 32) / 256, 256, 0, stream>>>(W, bias, acc);
    vq_argmax_kernel<<<N_TOK / ROWS_PER_BLOCK, 256, 0, stream>>>(X, W, bias, idx);
    vq_gather_kernel<<<N_TOK / WAVES, 256, 0, stream>>>(X, W, idx, out, acc);
    vq_finalize_kernel<<<1, 1, 0, stream>>>(acc, out + (size_t)N_TOK * DIM);
}